// MultiHeadAttention_14972255994381
// MI455X (gfx1250) — compile-verified
//
#include <hip/hip_runtime.h>

// MI455X / gfx1250 — wave32, WMMA f32 16x16x4.
typedef float v2f __attribute__((ext_vector_type(2)));
typedef float v8f __attribute__((ext_vector_type(8)));

#define WMMA_F32(a, b, c) \
  __builtin_amdgcn_wmma_f32_16x16x4_f32(false, (a), false, (b), (short)0, (c), false, false)

constexpr int Bb = 2, Ss = 2048, Ee = 1024, Hh = 16, DKk = 64;
constexpr int HD = Hh * DKk;  // 1024

// ---------------------------------------------------------------------------
// Generic 1024-K GEMM:  Y = X(rows,1024) @ W(1024,1024) + bias
// One wave computes a 32x64 output tile (2 M-tiles x 4 N-tiles) so each
// W-fragment load feeds two WMMAs (10 loads / 8 wmma per k-step).
// mode 0: scatter into (B,H,S,DK); mode 1: row-major.
// ---------------------------------------------------------------------------
__global__ __launch_bounds__(128) void gemm1024_kernel(
    const float* __restrict__ X, const float* __restrict__ W,
    const float* __restrict__ bias, float* __restrict__ Y, int mode) {
  const int lane = threadIdx.x & 31;
  const int wv   = threadIdx.x >> 5;
  const int hf   = (lane >> 4) & 1;
  const int l16  = lane & 15;

  const int waveId  = blockIdx.x * 4 + wv;   // 2048 waves total
  const int rowTile = waveId >> 4;           // 0..127  (rows/32)
  const int colStrp = waveId & 15;           // 0..15   (1024/64)
  const int row0 = rowTile * 32;
  const int col0 = colStrp * 64;

  v8f acc[2][4];
#pragma unroll
  for (int nt = 0; nt < 4; ++nt) {
    const float bv = bias[col0 + nt * 16 + l16];
#pragma unroll
    for (int mt = 0; mt < 2; ++mt)
#pragma unroll
      for (int r = 0; r < 8; ++r) acc[mt][nt][r] = bv;
  }

  const float* xrow0 = X + (size_t)(row0 + l16) * Ee;
  const float* xrow1 = xrow0 + (size_t)16 * Ee;
  for (int k = 0; k < Ee; k += 4) {
    const int kk = k + 2 * hf;
    v2f a0, a1;
    a0[0] = xrow0[kk];
    a0[1] = xrow0[kk + 1];
    a1[0] = xrow1[kk];
    a1[1] = xrow1[kk + 1];
    const float* w0 = W + (size_t)kk * HD + col0;
    const float* w1 = w0 + HD;
#pragma unroll
    for (int nt = 0; nt < 4; ++nt) {
      v2f b;
      b[0] = w0[nt * 16 + l16];
      b[1] = w1[nt * 16 + l16];
      acc[0][nt] = WMMA_F32(a0, b, acc[0][nt]);
      acc[1][nt] = WMMA_F32(a1, b, acc[1][nt]);
    }
  }

#pragma unroll
  for (int mt = 0; mt < 2; ++mt) {
#pragma unroll
    for (int nt = 0; nt < 4; ++nt) {
      const int col = col0 + nt * 16 + l16;
#pragma unroll
      for (int r = 0; r < 8; ++r) {
        const int row = row0 + mt * 16 + r + 8 * hf;
        size_t dst;
        if (mode == 0) {  // (B,H,S,DK) scatter: row = b*S+s, col = h*64+d
          const int b = row >> 11, s = row & (Ss - 1);
          const int h = col >> 6, d = col & 63;
          dst = (((size_t)(b * Hh + h)) * Ss + s) * DKk + d;
        } else {
          dst = (size_t)row * Ee + col;
        }
        Y[dst] = acc[mt][nt][r];
      }
    }
  }
}

// ---------------------------------------------------------------------------
// Fused attention: per wave, one (b,h,16-query tile).
//  - scores tile = (Q/8) @ K^T  (16 wmma), mask, write masked scores once
//  - online softmax (m,l per row) + P@V (16 wmma) accumulated in registers
//  - P transposed C-layout -> A-layout through LDS
// ---------------------------------------------------------------------------
__global__ __launch_bounds__(128) void attn_kernel(
    const float* __restrict__ Qh, const float* __restrict__ Kh,
    const float* __restrict__ Vh, const int* __restrict__ mask,
    float* __restrict__ scores, float* __restrict__ concat) {
  __shared__ float pT[4][16][17];  // per-wave transpose slot, padded
  const int lane = threadIdx.x & 31;
  const int wv   = threadIdx.x >> 5;
  const int hf   = (lane >> 4) & 1;
  const int l16  = lane & 15;

  const int waveId = blockIdx.x * 4 + wv;  // 4096 waves = B*H*(S/16)
  const int qTile  = waveId & 127;
  const int bh     = waveId >> 7;  // 0..31
  const int b      = bh >> 4;
  const int h      = bh & 15;
  const int q0     = qTile * 16;

  const size_t headOff = (size_t)bh * Ss * DKk;
  const float* Qb = Qh + headOff;
  const float* Kb = Kh + headOff;
  const float* Vb = Vh + headOff;

  // Q fragments for 16 k-steps, pre-scaled by 1/sqrt(DK)
  v2f aQ[16];
  {
    const float* qrow = Qb + (size_t)(q0 + l16) * DKk + 2 * hf;
#pragma unroll
    for (int st = 0; st < 16; ++st) {
      aQ[st][0] = qrow[4 * st] * 0.125f;
      aQ[st][1] = qrow[4 * st + 1] * 0.125f;
    }
  }

  v8f acc[4];
  float m[8], l[8];
#pragma unroll
  for (int nt = 0; nt < 4; ++nt)
#pragma unroll
    for (int r = 0; r < 8; ++r) acc[nt][r] = 0.0f;
#pragma unroll
  for (int r = 0; r < 8; ++r) { m[r] = -3.0e38f; l[r] = 0.0f; }

  float* scoreBase = scores + (((size_t)bh) * Ss + q0) * Ss;

  for (int kt = 0; kt < Ss / 16; ++kt) {
    const int k0 = kt * 16;
    const float* krow = Kb + (size_t)(k0 + l16) * DKk + 2 * hf;
    if (kt + 1 < Ss / 16) __builtin_prefetch(krow + 16 * DKk, 0, 1);

    // ---- scores tile: (Q/8) @ K^T ----
    v8f s;
#pragma unroll
    for (int r = 0; r < 8; ++r) s[r] = 0.0f;
#pragma unroll
    for (int st = 0; st < 16; ++st) {
      v2f bk;
      bk[0] = krow[4 * st];
      bk[1] = krow[4 * st + 1];
      s = WMMA_F32(aQ[st], bk, s);
    }

    // ---- mask (per key column) + write masked scores ----
    const int mv = mask[b * Ss + k0 + l16];
    if (mv == 0) {
#pragma unroll
      for (int r = 0; r < 8; ++r) s[r] = -1.0e9f;
    }
#pragma unroll
    for (int r = 0; r < 8; ++r)
      scoreBase[(size_t)(r + 8 * hf) * Ss + k0 + l16] = s[r];

    // ---- online softmax (rows live in 16-lane halves) ----
    float p[8], resc[8];
#pragma unroll
    for (int r = 0; r < 8; ++r) {
      float t = s[r];
      t = fmaxf(t, __shfl_xor(t, 8, 32));
      t = fmaxf(t, __shfl_xor(t, 4, 32));
      t = fmaxf(t, __shfl_xor(t, 2, 32));
      t = fmaxf(t, __shfl_xor(t, 1, 32));
      const float mnew = fmaxf(m[r], t);
      resc[r] = __expf(m[r] - mnew);
      const float pv = __expf(s[r] - mnew);
      float sum = pv;
      sum += __shfl_xor(sum, 8, 32);
      sum += __shfl_xor(sum, 4, 32);
      sum += __shfl_xor(sum, 2, 32);
      sum += __shfl_xor(sum, 1, 32);
      l[r] = l[r] * resc[r] + sum;
      m[r] = mnew;
      p[r] = pv;
    }
#pragma unroll
    for (int nt = 0; nt < 4; ++nt)
#pragma unroll
      for (int r = 0; r < 8; ++r) acc[nt][r] *= resc[r];

    // ---- transpose P (C-layout -> A-layout) through LDS ----
#pragma unroll
    for (int r = 0; r < 8; ++r) pT[wv][r + 8 * hf][l16] = p[r];
    v2f aP[4];
#pragma unroll
    for (int st = 0; st < 4; ++st) {
      aP[st][0] = pT[wv][l16][4 * st + 2 * hf];
      aP[st][1] = pT[wv][l16][4 * st + 2 * hf + 1];
    }

    // ---- P @ V ----
#pragma unroll
    for (int st = 0; st < 4; ++st) {
      const float* v0 = Vb + (size_t)(k0 + 4 * st + 2 * hf) * DKk;
      const float* v1 = v0 + DKk;
#pragma unroll
      for (int nt = 0; nt < 4; ++nt) {
        v2f bv;
        bv[0] = v0[nt * 16 + l16];
        bv[1] = v1[nt * 16 + l16];
        acc[nt] = WMMA_F32(aP[st], bv, acc[nt]);
      }
    }
  }

  // ---- normalize and write concat (B,S,H*DK) ----
  float inv[8];
#pragma unroll
  for (int r = 0; r < 8; ++r) inv[r] = 1.0f / l[r];
  float* cb = concat + (size_t)b * Ss * HD;
#pragma unroll
  for (int nt = 0; nt < 4; ++nt) {
    const int col = h * DKk + nt * 16 + l16;
#pragma unroll
    for (int r = 0; r < 8; ++r) {
      const int q = q0 + r + 8 * hf;
      cb[(size_t)q * HD + col] = acc[nt][r] * inv[r];
    }
  }
}

// ---------------------------------------------------------------------------
extern "C" void kernel_launch(void* const* d_in, const int* in_sizes, int n_in,
                              void* d_out, int out_size, void* d_ws, size_t ws_size,
                              hipStream_t stream) {
  const float* q    = (const float*)d_in[0];
  const float* k    = (const float*)d_in[1];
  const float* v    = (const float*)d_in[2];
  const int*   mask = (const int*)d_in[3];
  const float* Wq = (const float*)d_in[4];
  const float* bq = (const float*)d_in[5];
  const float* Wk = (const float*)d_in[6];
  const float* bk = (const float*)d_in[7];
  const float* Wv = (const float*)d_in[8];
  const float* bv = (const float*)d_in[9];
  const float* Wo = (const float*)d_in[10];
  const float* bo = (const float*)d_in[11];

  float* out    = (float*)d_out;                  // (B,S,E)
  float* scores = out + (size_t)Bb * Ss * Ee;     // (B,H,S,S)

  const size_t headElems = (size_t)Bb * Hh * Ss * DKk;  // 4M floats
  float* Qh     = (float*)d_ws;
  float* Kh     = Qh + headElems;
  float* Vh     = Kh + headElems;
  float* concat = Vh + headElems;                 // (B,S,H*DK)

  dim3 blk(128);
  dim3 gemmGrd(512);  // 2048 waves, 32x64 tile each
  dim3 attnGrd(1024); // 4096 waves
  gemm1024_kernel<<<gemmGrd, blk, 0, stream>>>(q, Wq, bq, Qh, 0);
  gemm1024_kernel<<<gemmGrd, blk, 0, stream>>>(k, Wk, bk, Kh, 0);
  gemm1024_kernel<<<gemmGrd, blk, 0, stream>>>(v, Wv, bv, Vh, 0);
  attn_kernel<<<attnGrd, blk, 0, stream>>>(Qh, Kh, Vh, mask, scores, concat);
  gemm1024_kernel<<<gemmGrd, blk, 0, stream>>>(concat, Wo, bo, out, 1);
}